// LinearModel_58626303590600
// MI455X (gfx1250) — compile-verified
//
#include <hip/hip_runtime.h>
#include <stdint.h>

#define B_ROWS 256
#define L_LEN  4096
#define D_DIM  40
#define UNITS  101
#define PADVF  39.0f
#define TABLE_ELEMS (UNITS * D_DIM)      // 4040 floats = 16160 bytes
#define TOK_PER_BLK 512
#define TOKENS (B_ROWS * L_LEN)          // 1,048,576

typedef int   v4i __attribute__((ext_vector_type(4)));
typedef int   v8i __attribute__((ext_vector_type(8)));
typedef float f4  __attribute__((ext_vector_type(4)));

// ---------------- Kernel A: per-unit scale + argmax (tiny) ----------------
__global__ void precompute_table(const float* __restrict__ W,
                                 float* __restrict__ Wn,
                                 int* __restrict__ argLab) {
  int u = threadIdx.x;
  if (u >= UNITS) return;
  float row[D_DIM];
  float ss = 0.f;
  float best = -3.402823466e+38f;
  int bi = 0;
#pragma unroll
  for (int d = 0; d < D_DIM; ++d) {
    float v = W[u * D_DIM + d];
    row[d] = v;
    ss += v * v;
    if (v > best) { best = v; bi = d; }   // strict > keeps first occurrence (jnp.argmax)
  }
  float nrm = sqrtf(ss);
  float scale = fminf(1.0f, 1.0f / fmaxf(nrm, 1e-12f));
#pragma unroll
  for (int d = 0; d < D_DIM; ++d) Wn[u * D_DIM + d] = row[d] * scale;
  argLab[u] = bi;
}

// ------------- TDM: async DMA 16160B table (global -> LDS) ----------------
__device__ __forceinline__ void tdm_load_table(const float* gsrc, void* lds_dst) {
  uint64_t ga = (uint64_t)(uintptr_t)gsrc;
  uint32_t lo = (uint32_t)(uintptr_t)lds_dst;     // low 32 bits = LDS byte offset
  v4i g0;
  g0.x = 1;                                       // count=1, user descriptor
  g0.y = (int)lo;                                 // lds_addr
  g0.z = (int)(uint32_t)(ga & 0xffffffffull);     // global_addr[31:0]
  g0.w = (int)(uint32_t)(((ga >> 32) & 0x01ffffffull) | 0x80000000u); // addr[56:32] | type=2
  v8i g1;
  g1[0] = (2 << 16);                              // data_size = 4 bytes; no mask/pad/iter
  g1[1] = (TABLE_ELEMS & 0xffff) << 16;           // tensor_dim0 lo
  g1[2] = ((TABLE_ELEMS >> 16) & 0xffff) | (1 << 16); // tensor_dim0 hi | tensor_dim1=1
  g1[3] = (TABLE_ELEMS & 0xffff) << 16;           // tile_dim0 = 4040
  g1[4] = 1;                                      // tile_dim1 = 1, tile_dim2 = 0
  g1[5] = TABLE_ELEMS;                            // tensor_dim0_stride lo
  g1[6] = (TABLE_ELEMS & 0xffff) << 16;           // stride0 hi=0 | tensor_dim1_stride lo
  g1[7] = 0;
  asm volatile("tensor_load_to_lds %0, %1" :: "s"(g0), "s"(g1) : "memory");
}

// -------- Kernel B: probs[t][*] = scaled_table[x[t]][*]  (BW-bound) -------
__global__ void __launch_bounds__(256)
gather_probs(const int* __restrict__ x, const float* __restrict__ Wn,
             float* __restrict__ probs) {
  __shared__ __align__(16) float tab[TABLE_ELEMS];
  __shared__ int toks[TOK_PER_BLK];
  const int tid = threadIdx.x;
  const long long tokBase = (long long)blockIdx.x * TOK_PER_BLK;

  if (tid < 32) {                        // uniform per-wave: only wave 0 issues TDM
    tdm_load_table(Wn, (void*)tab);
    __builtin_amdgcn_s_wait_tensorcnt(0);
  }
  toks[tid]       = x[tokBase + tid];
  toks[tid + 256] = x[tokBase + tid + 256];
  __syncthreads();

  f4* __restrict__ outp = (f4*)(probs + tokBase * D_DIM);
  const f4* tabv = (const f4*)tab;
#pragma unroll
  for (int k = 0; k < 20; ++k) {         // 5120 float4 = 512 tokens * 10
    unsigned q = (unsigned)tid + k * 256u;
    unsigned token = q / 10u;            // each float4 lies inside one token's 40-float row
    unsigned d4 = q - token * 10u;
    int unit = toks[token];
    f4 v = tabv[(unsigned)unit * 10u + d4];
    __builtin_nontemporal_store(v, &outp[q]);   // 167.8MB write-only stream: keep out of L2
  }
}

// ---- Kernel C: labels / run-length encode (labels_ded, inv) per row ------
__global__ void __launch_bounds__(256)
rle_labels(const int* __restrict__ x, const int* __restrict__ argLab,
           float* __restrict__ labels_out, float* __restrict__ ded_out,
           float* __restrict__ inv_out) {
  __shared__ int argl[UNITS];
  __shared__ int cnt[256];
  const int t = threadIdx.x;
  const long long rbase = (long long)blockIdx.x * L_LEN;
  if (t < UNITS) argl[t] = argLab[t];
  __syncthreads();

  const int p0 = t * 16;
  int xv[16];
  const v4i* xp = (const v4i*)(x + rbase + p0);
#pragma unroll
  for (int j = 0; j < 4; ++j) {
    v4i v = xp[j];
    xv[j*4+0]=v.x; xv[j*4+1]=v.y; xv[j*4+2]=v.z; xv[j*4+3]=v.w;
  }
  int lab[16];
#pragma unroll
  for (int j = 0; j < 16; ++j) lab[j] = argl[xv[j]];

  int prev = (t == 0) ? -1 : argl[x[rbase + p0 - 1]];  // -1 forces keep at pos 0

  int c = 0, keepm = 0;
#pragma unroll
  for (int j = 0; j < 16; ++j) {
    int pv = (j == 0) ? prev : lab[j-1];
    int k = (lab[j] != pv) ? 1 : 0;
    keepm |= k << j;
    c += k;
  }

  // block-wide inclusive scan of per-thread keep counts
  cnt[t] = c;
  __syncthreads();
  for (int off = 1; off < 256; off <<= 1) {
    int v = cnt[t];
    int add = (t >= off) ? cnt[t - off] : 0;
    __syncthreads();
    cnt[t] = v + add;
    __syncthreads();
  }
  int run = cnt[t] - c;            // exclusive prefix of keeps
  const int total = cnt[255];      // total runs in this row

  float lf[16], invf[16];
#pragma unroll
  for (int j = 0; j < 16; ++j) {
    if ((keepm >> j) & 1) {
      ded_out[rbase + run] = (float)lab[j];   // scatter into [0, total)
      run++;
    }
    lf[j]   = (float)lab[j];
    invf[j] = (float)(run - 1);
  }
  // PAD fill [total, L) — disjoint from the scatter region, so no WAW hazard
#pragma unroll
  for (int j = 0; j < 16; ++j) {
    int pos = p0 + j;
    if (pos >= total) ded_out[rbase + pos] = PADVF;
  }

  f4* lp = (f4*)(labels_out + rbase + p0);
  f4* ip = (f4*)(inv_out + rbase + p0);
#pragma unroll
  for (int j = 0; j < 4; ++j) {
    f4 lv = { lf[j*4],   lf[j*4+1],   lf[j*4+2],   lf[j*4+3] };
    f4 iv = { invf[j*4], invf[j*4+1], invf[j*4+2], invf[j*4+3] };
    __builtin_nontemporal_store(lv, &lp[j]);    // write-only streams
    __builtin_nontemporal_store(iv, &ip[j]);
  }
}

// --------------------------------------------------------------------------
extern "C" void kernel_launch(void* const* d_in, const int* in_sizes, int n_in,
                              void* d_out, int out_size, void* d_ws, size_t ws_size,
                              hipStream_t stream) {
  (void)in_sizes; (void)n_in; (void)out_size; (void)ws_size;
  const int*   x = (const int*)d_in[0];          // [256,4096] int32
  const float* W = (const float*)d_in[1];        // [101,40]  f32

  float* Wn     = (float*)d_ws;                                   // 4040 f32
  int*   argLab = (int*)((char*)d_ws + TABLE_ELEMS * sizeof(float)); // 101 i32

  float* out    = (float*)d_out;
  float* probs  = out;                                            // 256*4096*40
  float* labels = out + (long long)TOKENS * D_DIM;                // 1,048,576
  float* ded    = labels + TOKENS;                                // 1,048,576
  float* inv    = ded + TOKENS;                                   // 1,048,576

  precompute_table<<<1, 128, 0, stream>>>(W, Wn, argLab);
  gather_probs<<<TOKENS / TOK_PER_BLK, 256, 0, stream>>>(x, Wn, probs);
  rle_labels<<<B_ROWS, 256, 0, stream>>>(x, argLab, labels, ded, inv);
}